// QuantizeLayer_78786880078104
// MI455X (gfx1250) — compile-verified
//
#include <hip/hip_runtime.h>
#include <hip/hip_bf16.h>
#include <float.h>

// Problem constants (match reference)
#define B_DIM 8
#define T_DIM 512
#define E_DIM 128
#define N_CODE 256
#define BT (B_DIM * T_DIM)            // 4096 tokens
#define BTE (BT * E_DIM)              // 524288 output elems for z_q_st
#define BETA 0.25f

// Tiling
#define NC 64                          // codewords staged per chunk
#define NCH (N_CODE / NC)              // 4 chunks
#define STR 130                        // LDS row stride in dwords (conflict-free, 8B aligned)
#define WAVES 8                        // 256-thread block = 8 wave32
#define TPW 2                          // tokens per wave
#define TOK_PER_BLK (WAVES * TPW)      // 16
#define NBLK (BT / TOK_PER_BLK)        // 256 blocks

#if __has_builtin(__builtin_amdgcn_global_load_async_to_lds_b64)
#define HAS_ASYNC_LDS 1
#endif
#if __has_builtin(__builtin_amdgcn_wmma_f32_16x16x4_f32)
#define HAS_WMMA4 1
#endif

typedef __attribute__((ext_vector_type(2))) float v2f;
typedef __attribute__((ext_vector_type(8))) float v8f;
typedef __attribute__((ext_vector_type(2))) int   v2i;

#ifdef HAS_ASYNC_LDS
typedef __attribute__((address_space(1))) v2i* gas_b64_ptr;  // global <2 x i32>*
typedef __attribute__((address_space(3))) v2i* las_b64_ptr;  // LDS    <2 x i32>*
#endif

__device__ __forceinline__ void async_copy8(const float* g, float* l) {
#ifdef HAS_ASYNC_LDS
  __builtin_amdgcn_global_load_async_to_lds_b64(
      (gas_b64_ptr)(void*)g, (las_b64_ptr)(void*)l, 0, 0);
#else
  *(float2*)l = *(const float2*)g;
#endif
}

__device__ __forceinline__ void wait_async_lds() {
#ifdef HAS_ASYNC_LDS
#if __has_builtin(__builtin_amdgcn_s_wait_asynccnt)
  __builtin_amdgcn_s_wait_asynccnt(0);
#else
  asm volatile("s_wait_asynccnt 0" ::: "memory");
#endif
#endif
}

__device__ __forceinline__ void argmin_xor_reduce(float& d, int& i) {
#pragma unroll
  for (int off = 16; off > 0; off >>= 1) {
    float od = __shfl_xor(d, off, 32);
    int   oi = __shfl_xor(i, off, 32);
    // strict < keeps lowest index on ties (matches jnp.argmin first-min)
    if (od < d || (od == d && oi < i)) { d = od; i = oi; }
  }
}

__global__ __launch_bounds__(256) void vq_nearest_kernel(
    const float* __restrict__ x, const int* __restrict__ mask,
    const float* __restrict__ emb, float* __restrict__ out,
    float* __restrict__ part) {
  __shared__ __align__(16) float s_emb[NC * STR];                 // 33.3 KB
  __shared__ __align__(16) float s_x[WAVES][TPW][E_DIM];          // 8 KB
  __shared__ float s_red[WAVES];

  const int tid  = threadIdx.x;
  const int lane = tid & 31;
  const int wv   = tid >> 5;
  const int tok0 = blockIdx.x * TOK_PER_BLK + wv * TPW;

  // Stage this wave's two masked x rows into LDS (broadcast source for the
  // distance loop; same-address LDS reads are conflict-free broadcasts).
#pragma unroll
  for (int tk = 0; tk < TPW; ++tk) {
    const int t = tok0 + tk;
    float4 xv = ((const float4*)(x + (size_t)t * E_DIM))[lane];
    if (mask[t] != 0) xv = make_float4(0.f, 0.f, 0.f, 0.f);
    ((float4*)&s_x[wv][tk][0])[lane] = xv;
  }

  float bd0 = FLT_MAX, bd1 = FLT_MAX;
  int   bi0 = 0,       bi1 = 0;

  for (int ch = 0; ch < NCH; ++ch) {
    __syncthreads();  // previous chunk fully consumed before overwrite
    // Stage NC codewords: each wave copies NC/WAVES rows; per row each lane
    // moves 8 bytes twice (128 floats / 64 lane-slots) via async global->LDS.
#pragma unroll
    for (int r = 0; r < NC / WAVES; ++r) {
      const int row = wv * (NC / WAVES) + r;
      const float* src = emb + (size_t)(ch * NC + row) * E_DIM;
      float*       dst = s_emb + row * STR;
      async_copy8(src + lane * 2,      dst + lane * 2);
      async_copy8(src + 64 + lane * 2, dst + 64 + lane * 2);
    }
    wait_async_lds();
    __syncthreads();

    // Distances: lane owns codewords (ch*NC + lane) and (ch*NC + 32 + lane);
    // each ds_load_b64 of a codeword pair is reused for both tokens.
    float d00 = 0.f, d01 = 0.f, d10 = 0.f, d11 = 0.f;
    const float* xr0 = &s_x[wv][0][0];
    const float* xr1 = &s_x[wv][1][0];
    const float* cp0 = s_emb + lane * STR;
    const float* cp1 = s_emb + (lane + 32) * STR;
#pragma unroll 8
    for (int e = 0; e < E_DIM; e += 2) {
      const float2 ca = *(const float2*)(cp0 + e);
      const float2 cb = *(const float2*)(cp1 + e);
      const float2 xa = *(const float2*)(xr0 + e);
      const float2 xb = *(const float2*)(xr1 + e);
      d00 += fabsf(xa.x - ca.x) + fabsf(xa.y - ca.y);
      d01 += fabsf(xa.x - cb.x) + fabsf(xa.y - cb.y);
      d10 += fabsf(xb.x - ca.x) + fabsf(xb.y - ca.y);
      d11 += fabsf(xb.x - cb.x) + fabsf(xb.y - cb.y);
    }
    const int c0 = ch * NC + lane;
    const int c1 = ch * NC + 32 + lane;
    if (d00 < bd0) { bd0 = d00; bi0 = c0; }
    if (d01 < bd0) { bd0 = d01; bi0 = c1; }
    if (d10 < bd1) { bd1 = d10; bi1 = c0; }
    if (d11 < bd1) { bd1 = d11; bi1 = c1; }
  }

  argmin_xor_reduce(bd0, bi0);
  argmin_xor_reduce(bd1, bi1);

  float sq = 0.f;
#pragma unroll
  for (int tk = 0; tk < TPW; ++tk) {
    const int t  = tok0 + tk;
    const int bi = (tk == 0) ? bi0 : bi1;
    float4 o;
    if (mask[t] != 0) {
      o = make_float4(0.f, 0.f, 0.f, 0.f);   // masked: x=0, z_q=0, loss term 0
    } else {
      const float4 xv = ((const float4*)&s_x[wv][tk][0])[lane];
      const float4 zq = *(const float4*)(emb + (size_t)bi * E_DIM + lane * 4);
      o.x = xv.x + (zq.x - xv.x);
      o.y = xv.y + (zq.y - xv.y);
      o.z = xv.z + (zq.z - xv.z);
      o.w = xv.w + (zq.w - xv.w);
      const float dx = xv.x - zq.x, dy = xv.y - zq.y;
      const float dz = xv.z - zq.z, dw = xv.w - zq.w;
      sq += dx * dx + dy * dy + dz * dz + dw * dw;
    }
    ((float4*)(out + (size_t)t * E_DIM))[lane] = o;
  }

  // Deterministic block reduction of squared-error partial
#pragma unroll
  for (int off = 16; off > 0; off >>= 1) sq += __shfl_xor(sq, off, 32);
  if (lane == 0) s_red[wv] = sq;
  __syncthreads();
  if (tid == 0) {
    float tot = 0.f;
#pragma unroll
    for (int w = 0; w < WAVES; ++w) tot += s_red[w];
    part[blockIdx.x] = tot;
  }
}

// Single-wave final reduction of NBLK partials. Uses V_WMMA_F32_16X16X4_F32
// with an all-ones B matrix as a 64-element matrix-pipe row-sum (EXEC is all
// ones: one full wave, no divergence), then one shfl to combine halves.
__global__ __launch_bounds__(32) void vq_loss_reduce_kernel(
    const float* __restrict__ part, float* __restrict__ out) {
  const int lane = threadIdx.x;
  float p = 0.f;
#pragma unroll
  for (int i = lane; i < NBLK; i += 32) p += part[i];

  float total;
#ifdef HAS_WMMA4
  v2f a; a.x = p; a.y = 0.f;          // A[16x4]: row m gets p[m] (k=0) and p[m+16] (k=2)
  v2f b; b.x = 1.f; b.y = 1.f;        // B[4x16] = ones -> D[m][n] = p[m] + p[m+16]
  v8f c = {};
  v8f d = __builtin_amdgcn_wmma_f32_16x16x4_f32(
      false, a, false, b, (short)0, c, false, false);
  float s = d[0] + d[1] + d[2] + d[3] + d[4] + d[5] + d[6] + d[7];
  total = s + __shfl_xor(s, 16, 32);  // low-half rows + high-half rows
#else
  total = p;
#pragma unroll
  for (int off = 16; off > 0; off >>= 1) total += __shfl_xor(total, off, 32);
#endif

  if (lane == 0) {
    const float mse = total * (1.0f / (float)BTE);
    out[BTE]     = mse;          // embedding_loss
    out[BTE + 1] = BETA * mse;   // commitment_loss (same forward value * beta)
  }
}

extern "C" void kernel_launch(void* const* d_in, const int* in_sizes, int n_in,
                              void* d_out, int out_size, void* d_ws, size_t ws_size,
                              hipStream_t stream) {
  const float* x    = (const float*)d_in[0];   // (8,512,128) f32
  const int*   mask = (const int*)d_in[1];     // (8,512) bool->int
  const float* emb  = (const float*)d_in[2];   // (256,128) f32
  float* out  = (float*)d_out;                 // 524288 + 2
  float* part = (float*)d_ws;                  // NBLK partial sums

  vq_nearest_kernel<<<NBLK, WAVES * 32, 0, stream>>>(x, mask, emb, out, part);
  vq_loss_reduce_kernel<<<1, 32, 0, stream>>>(part, out);
}